// MixAttention_88192858456823
// MI455X (gfx1250) — compile-verified
//
#include <hip/hip_runtime.h>

typedef __attribute__((ext_vector_type(2))) float v2f;
typedef __attribute__((ext_vector_type(8))) float v8f;

#define N_ 8
#define L_ 8192
#define S_ 8192
#define H_ 8
#define D_ 32
#define EPS_ 1e-6f

__device__ __forceinline__ float fmap(float x) {
    // elu(x) + 1
    return x > 0.0f ? (x + 1.0f) : __expf(x);
}

__device__ __forceinline__ v8f wmma4f32(v2f a, v2f b, v8f c) {
    return __builtin_amdgcn_wmma_f32_16x16x4_f32(
        /*neg_a=*/false, a, /*neg_b=*/false, b,
        /*c_mod=*/(short)0, c, /*reuse_a=*/false, /*reuse_b=*/false);
}

__global__ void zero_ws_kernel(float* __restrict__ ws, int n) {
    int i = blockIdx.x * blockDim.x + threadIdx.x;
    if (i < n) ws[i] = 0.0f;
}

// Phase 1: per (n,h): gKV = K^T V (32x32), gKs = column sums of K (32)
// grid: (N_*H_, SCHUNKS), block: 256 (8 waves)
__global__ __launch_bounds__(256) void phase1_kernel(
        const float* __restrict__ Kg, const float* __restrict__ Vg,
        float* __restrict__ gKV, float* __restrict__ gKs) {
    __shared__ float skv[D_ * D_];
    __shared__ float sks[D_];

    const int nh   = blockIdx.x;
    const int n    = nh / H_;
    const int hd   = nh % H_;
    const int tid  = threadIdx.x;
    const int w    = tid >> 5;
    const int lane = tid & 31;
    const int hf   = lane >> 4;   // which half of the wave (K-group)
    const int mcol = lane & 15;   // M for A, N for B

    for (int i = tid; i < D_ * D_ + D_; i += 256) {
        if (i < D_ * D_) skv[i] = 0.0f;
        else             sks[i - D_ * D_] = 0.0f;
    }
    __syncthreads();

    v8f acc[2][2] = {};           // [dtile][vtile] 16x16 f32 tiles of KV
    float asum[2] = {0.0f, 0.0f}; // Ksum partials per dtile

    const int rows_per_wg   = S_ / gridDim.y;     // 1024 at gridDim.y==8
    const int rows_per_wave = rows_per_wg / 8;    // 128
    const int sb0 = blockIdx.y * rows_per_wg + w * rows_per_wave;

    const size_t base = ((size_t)n * S_) * (H_ * D_) + (size_t)hd * D_;

    for (int st = 0; st < rows_per_wave; st += 4) {
        const int s0 = sb0 + st + 2 * hf;
        const size_t r0 = base + (size_t)s0 * (H_ * D_);
        const size_t r1 = r0 + (H_ * D_);
        v2f a[2], b[2];
        #pragma unroll
        for (int dt = 0; dt < 2; ++dt) {
            // A = K^T : A[M=d, k=s]; coalesced b32 loads across lanes
            a[dt].x = fmap(Kg[r0 + dt * 16 + mcol]);
            a[dt].y = fmap(Kg[r1 + dt * 16 + mcol]);
            asum[dt] += a[dt].x + a[dt].y;
        }
        #pragma unroll
        for (int vt = 0; vt < 2; ++vt) {
            // B = V : B[k=s, N=v]
            b[vt].x = Vg[r0 + vt * 16 + mcol];
            b[vt].y = Vg[r1 + vt * 16 + mcol];
        }
        #pragma unroll
        for (int dt = 0; dt < 2; ++dt)
            #pragma unroll
            for (int vt = 0; vt < 2; ++vt)
                acc[dt][vt] = wmma4f32(a[dt], b[vt], acc[dt][vt]);
    }

    // reduce waves into LDS (ds_add_f32)
    #pragma unroll
    for (int dt = 0; dt < 2; ++dt) {
        #pragma unroll
        for (int vt = 0; vt < 2; ++vt)
            #pragma unroll
            for (int j = 0; j < 8; ++j) {
                const int d = dt * 16 + 8 * hf + j;   // C/D layout: M = j + 8*half
                const int v = vt * 16 + mcol;
                atomicAdd(&skv[d * D_ + v], acc[dt][vt][j]);
            }
        atomicAdd(&sks[dt * 16 + mcol], asum[dt]);
    }
    __syncthreads();

    float* kvDst = gKV + (size_t)nh * (D_ * D_);
    float* ksDst = gKs + (size_t)nh * D_;
    for (int i = tid; i < D_ * D_; i += 256) unsafeAtomicAdd(&kvDst[i], skv[i]);
    if (tid < D_) unsafeAtomicAdd(&ksDst[tid], sks[tid]);
}

// Phase 2: out = (Q @ KV) / (Q . Ksum + eps)
// grid: (N_*H_, 16), block: 256 (8 waves); each wave does 4 blocks of 16 rows
__global__ __launch_bounds__(256) void phase2_kernel(
        const float* __restrict__ Qg, const float* __restrict__ gKV,
        const float* __restrict__ gKs, float* __restrict__ Og) {
    const int nh   = blockIdx.x;
    const int n    = nh / H_;
    const int hd   = nh % H_;
    const int tid  = threadIdx.x;
    const int w    = tid >> 5;
    const int lane = tid & 31;
    const int hf   = lane >> 4;
    const int mcol = lane & 15;

    // B fragments (constant across all l): KV tiles + Ksum replicated over N
    const float* kv = gKV + (size_t)nh * (D_ * D_);
    const float* ks = gKs + (size_t)nh * D_;
    v2f bkv[2][8];
    v2f bks[8];
    #pragma unroll
    for (int kk = 0; kk < 8; ++kk) {
        const int k0 = 4 * kk + 2 * hf;
        #pragma unroll
        for (int vt = 0; vt < 2; ++vt) {
            bkv[vt][kk].x = kv[(k0    ) * D_ + vt * 16 + mcol];
            bkv[vt][kk].y = kv[(k0 + 1) * D_ + vt * 16 + mcol];
        }
        bks[kk].x = ks[k0];
        bks[kk].y = ks[k0 + 1];
    }

    const int rows_per_wg   = L_ / gridDim.y;     // 512 at gridDim.y==16
    const int rows_per_wave = rows_per_wg / 8;    // 64 -> 4 row-blocks of 16
    const int rb0 = blockIdx.y * rows_per_wg + w * rows_per_wave;
    const size_t base = ((size_t)n * L_) * (H_ * D_) + (size_t)hd * D_;

    for (int blk = 0; blk < rows_per_wave; blk += 16) {
        const int rb = rb0 + blk;
        v8f acc[2] = {};
        v8f accz = {};
        const size_t rowQ = base + (size_t)(rb + mcol) * (H_ * D_);
        #pragma unroll
        for (int kk = 0; kk < 8; ++kk) {
            const int k0 = 4 * kk + 2 * hf;
            v2f a;
            a.x = fmap(Qg[rowQ + k0]);
            a.y = fmap(Qg[rowQ + k0 + 1]);
            acc[0] = wmma4f32(a, bkv[0][kk], acc[0]);
            acc[1] = wmma4f32(a, bkv[1][kk], acc[1]);
            accz   = wmma4f32(a, bks[kk],    accz);
        }
        #pragma unroll
        for (int j = 0; j < 8; ++j) {
            const float z = 1.0f / (accz[j] + EPS_);
            const int row = rb + 8 * hf + j;
            const size_t o = base + (size_t)row * (H_ * D_);
            Og[o + mcol]      = acc[0][j] * z;   // coalesced 64B per half-wave
            Og[o + 16 + mcol] = acc[1][j] * z;
        }
    }
}

extern "C" void kernel_launch(void* const* d_in, const int* in_sizes, int n_in,
                              void* d_out, int out_size, void* d_ws, size_t ws_size,
                              hipStream_t stream) {
    const float* Q = (const float*)d_in[0];
    const float* K = (const float*)d_in[1];
    const float* V = (const float*)d_in[2];
    float* out = (float*)d_out;

    float* gKV = (float*)d_ws;                       // N*H*32*32 f32
    float* gKs = gKV + (size_t)N_ * H_ * D_ * D_;    // N*H*32 f32
    const int nws = N_ * H_ * D_ * D_ + N_ * H_ * D_;

    zero_ws_kernel<<<(nws + 255) / 256, 256, 0, stream>>>((float*)d_ws, nws);
    phase1_kernel<<<dim3(N_ * H_, 8), 256, 0, stream>>>(K, V, gKV, gKs);
    phase2_kernel<<<dim3(N_ * H_, 16), 256, 0, stream>>>(Q, gKV, gKs, out);
}